// RobustGCN_18047452578194
// MI455X (gfx1250) — compile-verified
//
#include <hip/hip_runtime.h>
#include <math.h>

// ---------------------------------------------------------------------------
// RobustGCN forward for MI455X (gfx1250, wave32).
// Dense GEMMs use V_WMMA_F32_16X16X4_F32 (fp32-exact matrix pipe).
// A-tile staging uses GLOBAL_LOAD_ASYNC_TO_LDS_B128 (ASYNCcnt) when the
// toolchain exposes the builtins; otherwise falls back to load+ds_store.
// SPMM is a deterministic CSR gather (no float atomics) with prefetch.
// ---------------------------------------------------------------------------

typedef __attribute__((ext_vector_type(2))) float v2f;
typedef __attribute__((ext_vector_type(8))) float v8f;
typedef __attribute__((ext_vector_type(4))) int v4i;

#ifndef __has_builtin
#define __has_builtin(x) 0
#endif

#if __has_builtin(__builtin_amdgcn_wmma_f32_16x16x4_f32)
#define HAVE_WMMA_F32_16X16X4 1
#else
#define HAVE_WMMA_F32_16X16X4 0
#endif

#if __has_builtin(__builtin_amdgcn_global_load_async_to_lds_b128) && \
    __has_builtin(__builtin_amdgcn_s_wait_asynccnt)
#define HAVE_ASYNC_LDS 1
#else
#define HAVE_ASYNC_LDS 0
#endif

__device__ __forceinline__ v8f mma_16x16x4_f32(v2f a, v2f b, v8f c) {
#if HAVE_WMMA_F32_16X16X4
  // 8 args: (neg_a, A, neg_b, B, c_mod, C, reuse_a, reuse_b)
  return __builtin_amdgcn_wmma_f32_16x16x4_f32(false, a, false, b, (short)0, c,
                                               false, false);
#else
  c[0] += a.x * b.x + a.y * b.y;  // placeholder; wmma=0 signals missing builtin
  return c;
#endif
}

// 16-byte global -> LDS copy (ASYNCcnt-tracked when builtins available).
// The builtin takes generic pointers to 16-byte vectors.
__device__ __forceinline__ void copy16_g2l(void* lds_dst, const void* gsrc) {
#if HAVE_ASYNC_LDS
  __builtin_amdgcn_global_load_async_to_lds_b128((v4i*)gsrc, (v4i*)lds_dst, 0,
                                                 0);
#else
  *(float4*)lds_dst = *(const float4*)gsrc;
#endif
}

__device__ __forceinline__ void async_copies_wait() {
#if HAVE_ASYNC_LDS
  __builtin_amdgcn_s_wait_asynccnt(0);
#endif
}

// ---------------------------------------------------------------------------
// Small utility kernels
// ---------------------------------------------------------------------------

__global__ void zero_u32_kernel(unsigned* __restrict__ p, int n) {
  int i = blockIdx.x * 256 + threadIdx.x;
  if (i < n) p[i] = 0u;
}

__global__ void deg_count_kernel(const int* __restrict__ row,
                                 unsigned* __restrict__ cnt, int E) {
  int e = blockIdx.x * 256 + threadIdx.x;
  if (e < E) atomicAdd(&cnt[row[e]], 1u);
}

// deg = cnt + 1 (self loop)  ->  ds = deg^-1/2, di = deg^-1
__global__ void deg_norm_kernel(const unsigned* __restrict__ cnt,
                                float* __restrict__ ds, float* __restrict__ di,
                                int n) {
  int i = blockIdx.x * 256 + threadIdx.x;
  if (i < n) {
    float d = (float)(cnt[i] + 1u);
    ds[i] = rsqrtf(d);
    di[i] = 1.0f / d;
  }
}

// Exclusive block scan of cnt -> rowptr, per-block totals -> bsums.
__global__ void scan_block_kernel(const unsigned* __restrict__ cnt,
                                  unsigned* __restrict__ rowptr,
                                  unsigned* __restrict__ bsums, int n) {
  __shared__ unsigned s[256];
  int i = blockIdx.x * 256 + threadIdx.x;
  unsigned v = (i < n) ? cnt[i] : 0u;
  s[threadIdx.x] = v;
  __syncthreads();
  for (int off = 1; off < 256; off <<= 1) {
    unsigned t = (threadIdx.x >= (unsigned)off) ? s[threadIdx.x - off] : 0u;
    __syncthreads();
    s[threadIdx.x] += t;
    __syncthreads();
  }
  if (i < n) rowptr[i] = s[threadIdx.x] - v;  // exclusive
  if (threadIdx.x == 255) bsums[blockIdx.x] = s[255];
}

__global__ void scan_bsums_kernel(unsigned* __restrict__ bsums, int nb) {
  if (blockIdx.x == 0 && threadIdx.x == 0) {
    unsigned acc = 0;
    for (int b = 0; b < nb; ++b) {
      unsigned t = bsums[b];
      bsums[b] = acc;
      acc += t;
    }
  }
}

__global__ void scan_add_kernel(unsigned* __restrict__ rowptr,
                                const unsigned* __restrict__ bsums, int n,
                                unsigned total) {
  int i = blockIdx.x * 256 + threadIdx.x;
  if (i < n) rowptr[i] += bsums[blockIdx.x];
  if (i == 0) rowptr[n] = total;
}

__global__ void csr_fill_kernel(const int* __restrict__ row,
                                const int* __restrict__ col,
                                const unsigned* __restrict__ rowptr,
                                unsigned* __restrict__ fill,
                                int* __restrict__ csr_col, int E) {
  int e = blockIdx.x * 256 + threadIdx.x;
  if (e < E) {
    int r = row[e];
    unsigned pos = rowptr[r] + atomicAdd(&fill[r], 1u);
    csr_col[pos] = col[e];
  }
}

// Sort each row segment -> deterministic summation order (rows avg ~16 edges).
__global__ void csr_sort_kernel(int* __restrict__ csr,
                                const unsigned* __restrict__ rowptr, int n) {
  int i = blockIdx.x * 256 + threadIdx.x;
  if (i >= n) return;
  int beg = (int)rowptr[i], end = (int)rowptr[i + 1];
  for (int a = beg + 1; a < end; ++a) {
    int key = csr[a];
    int b = a - 1;
    while (b >= beg && csr[b] > key) {
      csr[b + 1] = csr[b];
      --b;
    }
    csr[b + 1] = key;
  }
}

// ---------------------------------------------------------------------------
// Fused dual GEMM + activation epilogue (WMMA f32 16x16x4).
//   outM = elu(Am @ Wm + bm), outV = relu(Av @ Wv + bv)            (POST=false)
//   POST=true additionally: var += 1e-6; att = exp(-var);
//       outM = mean*att; outV = var*att*att
// Block: 256 threads (8 wave32 waves); 128 rows/block; wave owns 16 rows.
// LDS stride 36: 16B-aligned stores, conflict-free 16-row column reads.
// ---------------------------------------------------------------------------
template <int KTOT, int NCT, bool TWO_A, bool POST>
__global__ __launch_bounds__(256) void gemm_dual_kernel(
    const float* __restrict__ Am, const float* __restrict__ Av, int lda,
    const float* __restrict__ Wm, const float* __restrict__ Wv,
    const float* __restrict__ bm, const float* __restrict__ bv, int ncols,
    float* __restrict__ outM, float* __restrict__ outV, int ldo, int nrows) {
  __shared__ float As_m[128 * 36];
  __shared__ float As_v[(TWO_A ? 128 : 1) * 36];
  __shared__ float BTm[(NCT * 16) * 36];
  __shared__ float BTv[(NCT * 16) * 36];

  const int tid = threadIdx.x;
  const int lane = tid & 31;
  const int wid = tid >> 5;
  const int half = lane >> 4;  // 0: lanes 0-15, 1: lanes 16-31
  const int l16 = lane & 15;
  const int row0 = blockIdx.x * 128;
  const int wrow = wid * 16;

  v8f accM[NCT], accV[NCT];
#pragma unroll
  for (int ct = 0; ct < NCT; ++ct)
#pragma unroll
    for (int v = 0; v < 8; ++v) {
      accM[ct][v] = 0.f;
      accV[ct][v] = 0.f;
    }

  for (int k0 = 0; k0 < KTOT; k0 += 32) {
    // --- stage A (128 x 32) via async global->LDS, rows clamped at the tail
#pragma unroll
    for (int pp = 0; pp < 4; ++pp) {
      int rl = (tid >> 3) + pp * 32;  // 0..127
      int c4 = (tid & 7) * 4;         // 0..28
      int rg = row0 + rl;
      if (rg > nrows - 1) rg = nrows - 1;
      copy16_g2l(&As_m[rl * 36 + c4], Am + (size_t)rg * lda + k0 + c4);
      if constexpr (TWO_A) {
        copy16_g2l(&As_v[rl * 36 + c4], Av + (size_t)rg * lda + k0 + c4);
      }
    }
    // --- stage B transposed (zero-padded to NCT*16 cols) ---
    constexpr int NC = NCT * 16;
    for (int idx = tid; idx < 32 * NC; idx += 256) {
      int k = idx / NC, c = idx % NC;
      float wm = 0.f, wv = 0.f;
      if (c < ncols) {
        wm = Wm[(size_t)(k0 + k) * ncols + c];
        wv = Wv[(size_t)(k0 + k) * ncols + c];
      }
      BTm[c * 36 + k] = wm;
      BTv[c * 36 + k] = wv;
    }
    async_copies_wait();
    __syncthreads();

    // --- WMMA inner loop over this K slice ---
#pragma unroll
    for (int kk = 0; kk < 32; kk += 4) {
      // A frag 16x4: lane holds A[m][kk+2*half], A[m][kk+2*half+1]
      const int ai = (wrow + l16) * 36 + kk + 2 * half;
      v2f aM = *(const v2f*)(&As_m[ai]);
      v2f aV = aM;
      if constexpr (TWO_A) aV = *(const v2f*)(&As_v[ai]);
#pragma unroll
      for (int ct = 0; ct < NCT; ++ct) {
        // B frag 4x16: lane holds B[kk+2*half][n], B[kk+2*half+1][n]
        const int bi = (ct * 16 + l16) * 36 + kk + 2 * half;
        v2f bM = *(const v2f*)(&BTm[bi]);
        v2f bV = *(const v2f*)(&BTv[bi]);
        accM[ct] = mma_16x16x4_f32(aM, bM, accM[ct]);
        accV[ct] = mma_16x16x4_f32(aV, bV, accV[ct]);
      }
    }
    __syncthreads();
  }

  // --- epilogue: bias + ELU / ReLU (+ attention weighting) ---
#pragma unroll
  for (int ct = 0; ct < NCT; ++ct) {
    const int col = ct * 16 + l16;
    const bool cok = (col < ncols);
    const float bmv = cok ? bm[col] : 0.f;
    const float bvv = cok ? bv[col] : 0.f;
#pragma unroll
    for (int v = 0; v < 8; ++v) {
      const int rg = row0 + wrow + v + 8 * half;  // C/D row = vgpr + 8*half
      float hm = accM[ct][v] + bmv;
      float hv = accV[ct][v] + bvv;
      hm = (hm > 0.f) ? hm : expm1f(hm);  // elu
      hv = fmaxf(hv, 0.f);                // relu
      float om = hm, ov = hv;
      if constexpr (POST) {
        hv += 1e-6f;
        float att = expf(-hv);
        om = hm * att;
        ov = hv * att * att;
      }
      if (cok && rg < nrows) {
        outM[(size_t)rg * ldo + col] = om;
        outV[(size_t)rg * ldo + col] = ov;
      }
    }
  }
}

// ---------------------------------------------------------------------------
// CSR aggregation (one wave32 per node, lanes own columns {lane, lane+32}):
//   outM[i] = ds[i] * ( sum_j ds[j]*mpre[j] + ds[i]*mpre[i] )
//   outV[i] = di[i] * ( sum_j di[j]*vpre[j] + di[i]*vpre[i] )
// Neighbor rows are 256B coalesced; prefetch 4 edges ahead to hide HBM latency
// (this gather is the dominant memory traffic of the whole pipeline).
// ---------------------------------------------------------------------------
template <int COLS, int LDI, int LDO>
__global__ __launch_bounds__(256) void agg_kernel(
    const float* __restrict__ mpre, const float* __restrict__ vpre,
    const int* __restrict__ csr_col, const unsigned* __restrict__ rowptr,
    const float* __restrict__ ds, const float* __restrict__ di,
    float* __restrict__ outM, float* __restrict__ outV, int n) {
  const int lane = threadIdx.x & 31;
  const int wid = threadIdx.x >> 5;
  const int i = blockIdx.x * 8 + wid;
  if (i >= n) return;
  const int c0 = lane, c1 = lane + 32;
  const bool has1 = (c1 < COLS);

  float am0 = 0.f, am1 = 0.f, av0 = 0.f, av1 = 0.f;
  const unsigned beg = rowptr[i], end = rowptr[i + 1];
  for (unsigned e = beg; e < end; ++e) {
    if (e + 4 < end) {
      const int jp = csr_col[e + 4];
      __builtin_prefetch(mpre + (size_t)jp * LDI + c0, 0, 0);
      __builtin_prefetch(vpre + (size_t)jp * LDI + c0, 0, 0);
    }
    const int j = csr_col[e];
    const float ws = ds[j];
    const float wd = di[j];
    const float* mp = mpre + (size_t)j * LDI;
    const float* vp = vpre + (size_t)j * LDI;
    am0 += ws * mp[c0];
    av0 += wd * vp[c0];
    if (has1) {
      am1 += ws * mp[c1];
      av1 += wd * vp[c1];
    }
  }
  const float si = ds[i], dii = di[i];
  const float* mpi = mpre + (size_t)i * LDI;
  const float* vpi = vpre + (size_t)i * LDI;
  am0 += si * mpi[c0];
  av0 += dii * vpi[c0];
  outM[(size_t)i * LDO + c0] = si * am0;
  outV[(size_t)i * LDO + c0] = dii * av0;
  if (has1) {
    am1 += si * mpi[c1];
    av1 += dii * vpi[c1];
    outM[(size_t)i * LDO + c1] = si * am1;
    outV[(size_t)i * LDO + c1] = dii * av1;
  }
}

// ---------------------------------------------------------------------------
// Sampling + log-softmax (wave32 shuffle reduction over 40 classes)
// ---------------------------------------------------------------------------
__device__ __forceinline__ unsigned pcg_hash(unsigned x) {
  x = x * 747796405u + 2891336453u;
  unsigned w = ((x >> ((x >> 28u) + 4u)) ^ x) * 277803737u;
  return (w >> 22u) ^ w;
}

__device__ __forceinline__ float gauss_hash(unsigned idx) {
  unsigned a = pcg_hash(idx * 2u + 1u);
  unsigned b = pcg_hash(idx * 2u + 2u);
  float u1 = ((float)a + 1.0f) * 2.3283064365386963e-10f;  // (0,1]
  float u2 = (float)b * 2.3283064365386963e-10f;
  return sqrtf(fmaxf(-2.0f * logf(u1), 0.0f)) *
         cosf(6.283185307179586f * u2);
}

__global__ __launch_bounds__(256) void finalize_kernel(
    const float* __restrict__ mean, const float* __restrict__ var,
    float* __restrict__ out, int n) {
  const int lane = threadIdx.x & 31;
  const int wid = threadIdx.x >> 5;
  const int i = blockIdx.x * 8 + wid;
  if (i >= n) return;
  const int c0 = lane, c1 = lane + 32;

  float o0, o1 = -INFINITY;
  {
    float m = mean[(size_t)i * 40 + c0];
    float v = var[(size_t)i * 40 + c0];
    o0 = m + gauss_hash((unsigned)(i * 40 + c0)) * sqrtf(fmaxf(v, 1e-12f));
  }
  if (c1 < 40) {
    float m = mean[(size_t)i * 40 + c1];
    float v = var[(size_t)i * 40 + c1];
    o1 = m + gauss_hash((unsigned)(i * 40 + c1)) * sqrtf(fmaxf(v, 1e-12f));
  }
  float mx = fmaxf(o0, o1);
#pragma unroll
  for (int off = 16; off > 0; off >>= 1) mx = fmaxf(mx, __shfl_xor(mx, off, 32));
  float s = expf(o0 - mx) + ((c1 < 40) ? expf(o1 - mx) : 0.f);
#pragma unroll
  for (int off = 16; off > 0; off >>= 1) s += __shfl_xor(s, off, 32);
  const float l = mx + logf(s);
  out[(size_t)i * 40 + c0] = o0 - l;
  if (c1 < 40) out[(size_t)i * 40 + c1] = o1 - l;
}

// ---------------------------------------------------------------------------
// Launch
// ---------------------------------------------------------------------------
extern "C" void kernel_launch(void* const* d_in, const int* in_sizes, int n_in,
                              void* d_out, int out_size, void* d_ws,
                              size_t ws_size, hipStream_t stream) {
  const float* x = (const float*)d_in[0];
  const int* ei = (const int*)d_in[1];
  const float* Wm0 = (const float*)d_in[2];
  const float* bm0 = (const float*)d_in[3];
  const float* Wv0 = (const float*)d_in[4];
  const float* bv0 = (const float*)d_in[5];
  const float* Wm1 = (const float*)d_in[6];
  const float* bm1 = (const float*)d_in[7];
  const float* Wv1 = (const float*)d_in[8];
  const float* bv1 = (const float*)d_in[9];
  const float* Wm2 = (const float*)d_in[10];
  const float* bm2 = (const float*)d_in[11];
  const float* Wv2 = (const float*)d_in[12];
  const float* bv2 = (const float*)d_in[13];

  const int N = in_sizes[0] / 256;
  const int E = in_sizes[1] / 2;
  const int* erow = ei;
  const int* ecol = ei + E;

  char* p = (char*)d_ws;
  auto take = [&](size_t bytes) -> void* {
    void* r = (void*)p;
    p += (bytes + 255) & ~(size_t)255;
    return r;
  };
  unsigned* cnt = (unsigned*)take((size_t)N * 4);
  unsigned* rowptr = (unsigned*)take((size_t)(N + 1) * 4);
  unsigned* fillc = (unsigned*)take((size_t)N * 4);
  unsigned* bsums = (unsigned*)take(4096);
  float* ds = (float*)take((size_t)N * 4);
  float* di = (float*)take((size_t)N * 4);
  int* csr = (int*)take((size_t)E * 4);
  float* bufA = (float*)take((size_t)N * 64 * 4);
  float* bufB = (float*)take((size_t)N * 64 * 4);
  float* bufC = (float*)take((size_t)N * 64 * 4);
  float* bufD = (float*)take((size_t)N * 64 * 4);

  const int gN = (N + 255) / 256;
  const int gE = (E + 255) / 256;
  const int nb = gN;  // scan blocks over N entries
  const int gG = (N + 127) / 128;
  const int gW = (N + 7) / 8;

  // --- graph normalization: degrees, CSR build (deterministic) ---
  zero_u32_kernel<<<gN, 256, 0, stream>>>(cnt, N);
  zero_u32_kernel<<<gN, 256, 0, stream>>>(fillc, N);
  deg_count_kernel<<<gE, 256, 0, stream>>>(erow, cnt, E);
  deg_norm_kernel<<<gN, 256, 0, stream>>>(cnt, ds, di, N);
  scan_block_kernel<<<nb, 256, 0, stream>>>(cnt, rowptr, bsums, N);
  scan_bsums_kernel<<<1, 32, 0, stream>>>(bsums, nb);
  scan_add_kernel<<<nb, 256, 0, stream>>>(rowptr, bsums, N, (unsigned)E);
  csr_fill_kernel<<<gE, 256, 0, stream>>>(erow, ecol, rowptr, fillc, csr, E);
  csr_sort_kernel<<<gN, 256, 0, stream>>>(csr, rowptr, N);

  // --- layer 0: mean0 = elu(x@Wm0+bm0), var0 = relu(x@Wv0+bv0) ---
  gemm_dual_kernel<256, 4, false, false><<<gG, 256, 0, stream>>>(
      x, x, 256, Wm0, Wv0, bm0, bv0, 64, bufA, bufB, 64, N);

  // --- layer 1 dense + att; then CSR aggregation ---
  gemm_dual_kernel<64, 4, true, true><<<gG, 256, 0, stream>>>(
      bufA, bufB, 64, Wm1, Wv1, bm1, bv1, 64, bufC, bufD, 64, N);
  agg_kernel<64, 64, 64><<<gW, 256, 0, stream>>>(bufC, bufD, csr, rowptr, ds,
                                                 di, bufA, bufB, N);

  // --- layer 2 dense + att (40 cols, padded to 48); aggregation ---
  gemm_dual_kernel<64, 3, true, true><<<gG, 256, 0, stream>>>(
      bufA, bufB, 64, Wm2, Wv2, bm2, bv2, 40, bufC, bufD, 48, N);
  agg_kernel<40, 48, 40><<<gW, 256, 0, stream>>>(bufC, bufD, csr, rowptr, ds,
                                                 di, bufA, bufB, N);

  // --- sample + log_softmax ---
  finalize_kernel<<<gW, 256, 0, stream>>>(bufA, bufB, (float*)d_out, N);
}